// Model_24232205484608
// MI455X (gfx1250) — compile-verified
//
#include <hip/hip_runtime.h>
#include <stdint.h>

#define THRESH   0.5f
#define BLOCK    256          // 8 wave32s
#define VT       16           // elements per thread
#define TILE     (BLOCK * VT) // 4096 elements = 16 KB per tile
#define WAVES    (BLOCK / 32)
#define ROUNDS   4            // 4 x b128 loads per thread

typedef float v4f __attribute__((ext_vector_type(4)));

static constexpr uint64_t FLAG_AGG = 1ull << 62;
static constexpr uint64_t FLAG_INC = 1ull << 63;
static constexpr uint64_t VAL_MASK = (1ull << 62) - 1;

__device__ __forceinline__ int wave_incl_scan(int v, int lane) {
    #pragma unroll
    for (int d = 1; d < 32; d <<= 1) {
        int n = __shfl_up(v, (unsigned)d, 32);   // wave32 cross-lane
        if (lane >= d) v += n;
    }
    return v;
}

// Single-pass stream compaction with decoupled lookback (chained scan).
// Reads input exactly once (NT: bypass-friendly, L2 reserved for lookback
// state), writes only the kept elements (NT stores).
__global__ void __launch_bounds__(BLOCK)
compact_pass(const float* __restrict__ in,
             float* __restrict__ out,
             uint32_t* __restrict__ ticket,
             uint64_t* __restrict__ state,
             int numTiles, int N) {
    __shared__ uint32_t s_vbid;
    __shared__ uint32_t s_cnt[ROUNDS * WAVES]; // 32 entries: (round,wave) totals -> exclusive
    __shared__ uint32_t s_blockExcl;

    const int t    = threadIdx.x;
    const int lane = t & 31;
    const int w    = t >> 5;

    // Dynamic tile index: guarantees lookback forward progress on any scheduler.
    if (t == 0) s_vbid = atomicAdd(ticket, 1u);
    __syncthreads();
    const int vbid  = (int)s_vbid;
    const int base4 = vbid * (TILE / 4); // b128 index of tile start

    // ---- load tile: 4 fully-coalesced NT b128 loads per thread, all in flight ----
    const v4f* __restrict__ in4 = reinterpret_cast<const v4f*>(in);
    v4f v[ROUNDS];
    #pragma unroll
    for (int k = 0; k < ROUNDS; ++k) {
        int idx4 = base4 + k * BLOCK + t;
        int e0   = idx4 * 4;
        if (e0 + 3 < N) {
            v[k] = __builtin_nontemporal_load(in4 + idx4);
        } else { // ragged tail guard (unused for N = 2^26, kept for generality)
            v[k].x = (e0 + 0 < N) ? in[e0 + 0] : 0.0f;
            v[k].y = (e0 + 1 < N) ? in[e0 + 1] : 0.0f;
            v[k].z = (e0 + 2 < N) ? in[e0 + 2] : 0.0f;
            v[k].w = (e0 + 3 < N) ? in[e0 + 3] : 0.0f;
        }
    }

    // ---- intra-wave ranks per round ----
    int laneExcl[ROUNDS];
    #pragma unroll
    for (int k = 0; k < ROUNDS; ++k) {
        int cnt = (v[k].x >= THRESH) + (v[k].y >= THRESH) +
                  (v[k].z >= THRESH) + (v[k].w >= THRESH);
        int incl    = wave_incl_scan(cnt, lane);
        laneExcl[k] = incl - cnt;
        if (lane == 31) s_cnt[k * WAVES + w] = (uint32_t)incl; // wave total
    }
    __syncthreads();

    // ---- block scan of the 32 (round,wave) totals by wave 0, then lookback ----
    if (w == 0) {
        int c    = (int)s_cnt[lane];
        int incl = wave_incl_scan(c, lane);
        s_cnt[lane] = (uint32_t)(incl - c); // exclusive (round,wave) offsets
        uint32_t blockTotal = (uint32_t)__shfl(incl, 31, 32);
        if (lane == 0) {
            // publish aggregate so successors can pass through us
            __hip_atomic_store(&state[vbid], FLAG_AGG | (uint64_t)blockTotal,
                               __ATOMIC_RELEASE, __HIP_MEMORY_SCOPE_AGENT);
            uint64_t excl = 0;
            for (int p = vbid - 1; p >= 0;) {
                uint64_t s = __hip_atomic_load(&state[p], __ATOMIC_ACQUIRE,
                                               __HIP_MEMORY_SCOPE_AGENT);
                if (s & FLAG_INC)      { excl += s & VAL_MASK; break; }
                else if (s & FLAG_AGG) { excl += s & VAL_MASK; --p; }
                else                   { __builtin_amdgcn_s_sleep(1); }
            }
            __hip_atomic_store(&state[vbid], FLAG_INC | (excl + (uint64_t)blockTotal),
                               __ATOMIC_RELEASE, __HIP_MEMORY_SCOPE_AGENT);
            s_blockExcl = (uint32_t)excl;
        }
    }
    __syncthreads();

    // ---- scatter: lanes of a wave write a contiguous run -> coalesced NT stores ----
    const uint32_t blockExcl = s_blockExcl;
    #pragma unroll
    for (int k = 0; k < ROUNDS; ++k) {
        uint32_t p = blockExcl + s_cnt[k * WAVES + w] + (uint32_t)laneExcl[k];
        float x0 = v[k].x, x1 = v[k].y, x2 = v[k].z, x3 = v[k].w;
        if (x0 >= THRESH) __builtin_nontemporal_store(x0, out + p++);
        if (x1 >= THRESH) __builtin_nontemporal_store(x1, out + p++);
        if (x2 >= THRESH) __builtin_nontemporal_store(x2, out + p++);
        if (x3 >= THRESH) __builtin_nontemporal_store(x3, out + p++);
    }
}

// Zero-fill [count, N) with NT b128 stores and emit the count as out[N].
__global__ void __launch_bounds__(BLOCK)
zero_tail(const uint64_t* __restrict__ state, int numTiles,
          float* __restrict__ out, int N, int outSize) {
    const uint32_t count =
        (uint32_t)(state[numTiles - 1] & VAL_MASK); // final inclusive total
    const int tid    = blockIdx.x * blockDim.x + threadIdx.x;
    const int stride = gridDim.x * blockDim.x;

    if (tid == 0 && outSize > N) out[N] = (float)count;

    v4f* __restrict__ out4 = reinterpret_cast<v4f*>(out);
    const v4f zero4 = {0.f, 0.f, 0.f, 0.f};
    const int n4 = N >> 2;
    for (int j = tid; j < n4; j += stride) {
        int e = j << 2;
        if ((uint32_t)e >= count) {
            __builtin_nontemporal_store(zero4, out4 + j);
        } else if ((uint32_t)(e + 4) > count) {
            for (uint32_t q = count; q < (uint32_t)(e + 4); ++q) out[q] = 0.f;
        }
    }
    // ragged scalar tail (N % 4), none for N = 2^26
    int tail0 = n4 << 2;
    for (int e = tail0 + tid; e < N; e += stride)
        if ((uint32_t)e >= count) out[e] = 0.f;
}

extern "C" void kernel_launch(void* const* d_in, const int* in_sizes, int n_in,
                              void* d_out, int out_size, void* d_ws, size_t ws_size,
                              hipStream_t stream) {
    (void)n_in; (void)ws_size;
    const float* in  = (const float*)d_in[0];
    float*       out = (float*)d_out;
    const int N        = in_sizes[0];
    const int numTiles = (N + TILE - 1) / TILE; // 16384 for N = 2^26

    uint32_t* ticket = (uint32_t*)d_ws;
    uint64_t* state  = (uint64_t*)((char*)d_ws + 64);

    // Re-arm ticket + lookback flags every call (graph-capturable memset node).
    hipMemsetAsync(d_ws, 0, 64 + (size_t)numTiles * sizeof(uint64_t), stream);

    compact_pass<<<numTiles, BLOCK, 0, stream>>>(in, out, ticket, state,
                                                 numTiles, N);
    zero_tail<<<8192, BLOCK, 0, stream>>>(state, numTiles, out, N, out_size);
}